// MFVIConstituency_79774722556602
// MI455X (gfx1250) — compile-verified
//
#include <hip/hip_runtime.h>

#define S_DIM 192
#define B_DIM 8

typedef __attribute__((ext_vector_type(2))) float v2f;
typedef __attribute__((ext_vector_type(8))) float v8f;

__device__ __forceinline__ float sigmoid_f(float x) {
    return 1.0f / (1.0f + __expf(-x));
}

// P = sigmoid(s_con)   (initial iterate: q0 = s_con)
__global__ void mfvi_sig_init(const float* __restrict__ x,
                              float* __restrict__ p, int n) {
    int idx = blockIdx.x * blockDim.x + threadIdx.x;
    if (idx < n) p[idx] = sigmoid_f(x[idx]);
}

// One fixed-point pass:
//   Pnew[b,i,j] = sigmoid( s_con[b,i,j] + mask[b,i,j] *
//                  ( Σ_k s_bin[b,i,j,k]*Pold[b,j,k]
//                    - s_bin[b,i,j,i]*Pold[b,j,i]
//                    - (i!=j ? s_bin[b,i,j,j]*Pold[b,j,j] : 0) ) )
// Block = (b,i); 12 waves; wave w reduces rows j in [16w,16w+16) over k via
// V_WMMA_F32_16X16X4_F32 with an all-ones B (every accumulator column = row sum).
__global__ __launch_bounds__(384) void mfvi_update(
    const float* __restrict__ s_con,
    const float* __restrict__ s_bin,
    const unsigned char* __restrict__ mask,
    const float* __restrict__ Pold,
    float* __restrict__ Pnew)
{
    const int bi   = blockIdx.x;          // b*S + i
    const int b    = bi / S_DIM;
    const int i    = bi % S_DIM;
    const int lane = threadIdx.x & 31;
    const int wave = threadIdx.x >> 5;    // 0..11
    const int j0   = wave * 16;

    // 32-bit A-matrix 16x4 layout: lanes 0-15 -> M=0..15, K={0,1};
    // lanes 16-31 -> M=0..15, K={2,3}. Each lane loads a contiguous float2.
    const int m    = lane & 15;
    const int koff = (lane >> 4) << 1;    // 0 or 2
    const int j_a  = j0 + m;

    const float* sbRow = s_bin + ((((size_t)bi) * S_DIM + j_a) * S_DIM) + koff;
    const float* pRow  = Pold  + ((((size_t)b)  * S_DIM + j_a) * S_DIM) + koff;

    v8f c = {};
    const v2f bones = {1.0f, 1.0f};

    #pragma unroll 4
    for (int k0 = 0; k0 < S_DIM; k0 += 4) {
        v2f sv = *(const v2f*)(sbRow + k0);
        v2f pv = *(const v2f*)(pRow + k0);
        v2f a  = sv * pv;                 // Hadamard term for this 16x4 tile
        // D = A(16x4) x ones(4x16) + C  -> every column of D holds the row sums
        c = __builtin_amdgcn_wmma_f32_16x16x4_f32(
                /*neg_a=*/false, a, /*neg_b=*/false, bones,
                /*c_mod=*/(short)0, c, /*reuse_a=*/false, /*reuse_b=*/false);
    }

    // C/D layout: lanes 0-15 hold rows 0..7 in c[0..7]; lanes 16-31 hold rows 8..15.
    const int idx8 = lane & 7;
    float dot = c[0];
    #pragma unroll
    for (int t = 1; t < 8; ++t) dot = (idx8 == t) ? c[t] : dot;

    const bool active = (lane < 8) || (lane >= 16 && lane < 24);
    if (active) {
        const int j = j0 + ((lane >= 16) ? 8 : 0) + idx8;
        const size_t sbr = (((size_t)bi) * S_DIM + j) * S_DIM;
        const size_t pr  = (((size_t)b)  * S_DIM + j) * S_DIM;
        // algebraic masking: remove k==i and k==j contributions
        float ti = s_bin[sbr + i] * Pold[pr + i];
        float tj = (i != j) ? s_bin[sbr + j] * Pold[pr + j] : 0.0f;
        float dm = dot - ti - tj;
        const size_t o = ((size_t)bi) * S_DIM + j;
        float q = s_con[o] + (mask[o] ? dm : 0.0f);
        Pnew[o] = sigmoid_f(q);
    }
}

extern "C" void kernel_launch(void* const* d_in, const int* in_sizes, int n_in,
                              void* d_out, int out_size, void* d_ws, size_t ws_size,
                              hipStream_t stream) {
    const float*         s_con = (const float*)d_in[0];
    const float*         s_bin = (const float*)d_in[1];
    const unsigned char* mask  = (const unsigned char*)d_in[2]; // jax bool = 1 byte

    float* out = (float*)d_out;
    const int NP = B_DIM * S_DIM * S_DIM;      // 294912
    float* Pa = (float*)d_ws;                  // ping
    float* Pb = Pa + NP;                       // pong (total 2.36 MB of ws)

    mfvi_sig_init<<<(NP + 255) / 256, 256, 0, stream>>>(s_con, Pa, NP);

    const int nblk = B_DIM * S_DIM;            // 1536
    mfvi_update<<<nblk, 384, 0, stream>>>(s_con, s_bin, mask, Pa, Pb); // iter 1
    mfvi_update<<<nblk, 384, 0, stream>>>(s_con, s_bin, mask, Pb, Pa); // iter 2
    mfvi_update<<<nblk, 384, 0, stream>>>(s_con, s_bin, mask, Pa, out); // iter 3 -> sigmoid(q3)
}